// ConvQuad2D_57552561767075
// MI455X (gfx1250) — compile-verified
//
#include <hip/hip_runtime.h>

typedef __attribute__((ext_vector_type(2))) float v2f;
typedef __attribute__((ext_vector_type(4))) float v4f;
typedef __attribute__((ext_vector_type(8))) float v8f;

#define KW        5
#define PATCH     25            // K*K
#define JPAD      28            // j padded to 7 k-blocks of 4
#define NKB       7             // k-blocks per (d,i)
#define C_IN      4
#define F_OUT     16
#define HH        250
#define WW        250
#define BB        8
#define NPIX      (BB*HH*WW)    // 500000 (divisible by 16)
#define NTILES    (NPIX/16)     // 31250
#define WPB       16            // waves per block (512 threads) -> 4 waves/SIMD
#define NTHREADS  (WPB*32)
#define NQTILES   (C_IN*PATCH*NKB)   // 700 quad B-tiles
#define NLTILES   (C_IN*NKB)         // 28 linear B-tiles
#define NWTILES   (NQTILES+NLTILES)  // 728
#define WFLOATS   (NWTILES*64)       // 46592 floats of swizzled weights
#define PATF      (JPAD*C_IN)        // 112 floats per pixel patch
#define LDS_FLOATS (WFLOATS + WPB*16*PATF)  // 46592 + 28672 = 75264 -> 301,056 B (< 320KB WGP)

// ---------------------------------------------------------------------------
// Prep: swizzle quad_kernel (4,25,25,16) and kernel (5,5,4,16) into the
// V_WMMA_F32_16X16X4_F32 B-matrix lane layout, one 4x16 tile per k-block:
//   tile t, lane l, elem v (float2): B row K = v + 2*(l>=16), col N = l&15.
// Quad tile t = (d*25+i)*7+kb holds Wq[d,i, j=4kb+K, N]; padded j -> 0.
// Lin  tile t = 700 + d*7+kb holds kernel[i=4kb+K, d, N]; padded i -> 0.
// ---------------------------------------------------------------------------
__global__ void prep_weights(const float* __restrict__ kern,
                             const float* __restrict__ qk,
                             float* __restrict__ wbuf) {
    int idx = blockIdx.x * blockDim.x + threadIdx.x;
    if (idx >= WFLOATS) return;
    int t    = idx >> 6;
    int slot = idx & 63;
    int l    = slot >> 1;
    int v    = slot & 1;
    int n    = l & 15;
    int krow = v + ((l >= 16) ? 2 : 0);
    float val = 0.0f;
    if (t < NQTILES) {
        int kb = t % NKB;
        int di = t / NKB;              // d*25 + i
        int j  = 4 * kb + krow;
        if (j < PATCH) val = qk[(di * PATCH + j) * F_OUT + n];
    } else {
        int t2 = t - NQTILES;
        int kb = t2 % NKB;
        int d  = t2 / NKB;
        int i  = 4 * kb + krow;
        if (i < PATCH) val = kern[(i * C_IN + d) * F_OUT + n];
    }
    wbuf[idx] = val;
}

// ---------------------------------------------------------------------------
// Main: one wave = 16 output pixels. Per channel d:
//   stage1: S(16x16) = Pd(16x28) x Wq(28x16) per i  -> 7 WMMA f32 16x16x4
//   stage2: acc[g] += pat[pix(g),i,d] * S[g]        -> LDS broadcast + pk_fma
//   linear: acc = WMMA(Pd-block, Wlin-block, acc)   -> 7 WMMA (A reused)
// 16 waves/WG share the 186KB swizzled-weight LDS image -> 4 waves/SIMD.
// ---------------------------------------------------------------------------
extern __shared__ float smem[];

__global__ __launch_bounds__(NTHREADS)
void convquad_main(const float* __restrict__ x,
                   const float* __restrict__ bias,
                   const float* __restrict__ wbuf,
                   float* __restrict__ out) {
    const int tid  = threadIdx.x;
    const int lane = tid & 31;
    const int wave = tid >> 5;
    const int hi   = (lane >> 4) & 1;    // lane half selects K rows / pixel half
    const int n    = lane & 15;          // output feature column (and A pixel row)

    // ---- stage all swizzled weights into LDS (whole workgroup) ----
    {
        const v4f* src = (const v4f*)wbuf;
        v4f*       dst = (v4f*)smem;
        for (int e = tid; e < WFLOATS / 4; e += NTHREADS) dst[e] = src[e];
    }

    float* pat = smem + WFLOATS + wave * (16 * PATF);  // [pix][j(28)][d(4)]
    const int tile = blockIdx.x * WPB + wave;
    const bool active = (tile < NTILES);

    if (active) {
        const v4f z = {0.f, 0.f, 0.f, 0.f};
        // zero the padded j = 25..27 slots
        for (int e = lane; e < 16 * 3; e += 32) {
            int pix = e / 3, jj = PATCH + (e % 3);
            *(v4f*)(pat + pix * PATF + jj * C_IN) = z;
        }
        // extract SAME-padded 5x5 patches (4 channels contiguous -> b128)
        for (int e = lane; e < 16 * PATCH; e += 32) {
            int pix = e / PATCH, j = e % PATCH;
            int P  = tile * 16 + pix;
            int wc = P % WW;
            int hc = (P / WW) % HH;
            int bc = P / (WW * HH);
            int h2 = hc + (j / KW) - 2;
            int w2 = wc + (j % KW) - 2;
            v4f val = z;
            if (h2 >= 0 && h2 < HH && w2 >= 0 && w2 < WW)
                val = *(const v4f*)(x + (size_t)((bc * HH + h2) * WW + w2) * C_IN);
            *(v4f*)(pat + pix * PATF + j * C_IN) = val;
        }
    }
    __syncthreads();
    if (!active) return;

    const float bv = bias[n];
    v8f acc = {bv, bv, bv, bv, bv, bv, bv, bv};

    const float* patA = pat + n * PATF;        // A rows: pixel = lane&15
    const float* patS = pat + hi * 8 * PATF;   // stage-2: pixel = g + 8*hi

    #pragma unroll 1
    for (int d = 0; d < C_IN; ++d) {
        // A-blocks for channel d: lane<16 holds (j0, j0+1), lane>=16 (j0+2, j0+3)
        v2f A[NKB];
        #pragma unroll
        for (int kb = 0; kb < NKB; ++kb) {
            int j0 = 4 * kb + 2 * hi;
            A[kb].x = patA[j0 * C_IN + d];
            A[kb].y = patA[(j0 + 1) * C_IN + d];
        }
        const float* Bq = smem + (size_t)(d * PATCH) * NKB * 64 + lane * 2;
        #pragma unroll 1
        for (int i = 0; i < PATCH; ++i) {
            v8f s = {0.f, 0.f, 0.f, 0.f, 0.f, 0.f, 0.f, 0.f};
            const float* Bt = Bq + i * (NKB * 64);
            #pragma unroll
            for (int kb = 0; kb < NKB; ++kb) {
                v2f b = *(const v2f*)(Bt + kb * 64);
                s = __builtin_amdgcn_wmma_f32_16x16x4_f32(
                        false, A[kb], false, b, (short)0, s, false, false);
            }
            // stage 2: acc[g] += pat[g + 8*hi][i][d] * s[g]
            const float* m = patS + i * C_IN + d;
            #pragma unroll
            for (int g = 0; g < 8; ++g)
                acc[g] = __builtin_fmaf(m[g * PATF], s[g], acc[g]);
        }
        // linear conv term: reuse A-blocks with swizzled conv weights
        const float* Bl = smem + (size_t)(NQTILES + d * NKB) * 64 + lane * 2;
        #pragma unroll
        for (int kb = 0; kb < NKB; ++kb) {
            v2f b = *(const v2f*)(Bl + kb * 64);
            acc = __builtin_amdgcn_wmma_f32_16x16x4_f32(
                      false, A[kb], false, b, (short)0, acc, false, false);
        }
    }

    // store: pixel = tile*16 + hi*8 + g, feature = n
    float* op = out + (size_t)(tile * 16 + hi * 8) * F_OUT + n;
    #pragma unroll
    for (int g = 0; g < 8; ++g) op[g * F_OUT] = acc[g];
}

// ---------------------------------------------------------------------------
extern "C" void kernel_launch(void* const* d_in, const int* in_sizes, int n_in,
                              void* d_out, int out_size, void* d_ws, size_t ws_size,
                              hipStream_t stream) {
    const float* x    = (const float*)d_in[0];  // (8,250,250,4)
    const float* kern = (const float*)d_in[1];  // (5,5,4,16)
    const float* qk   = (const float*)d_in[2];  // (4,25,25,16)
    const float* bias = (const float*)d_in[3];  // (16,)
    float*       out  = (float*)d_out;          // (8,250,250,16)
    float*       wbuf = (float*)d_ws;           // 186,368 B of swizzled weights

    prep_weights<<<(WFLOATS + 255) / 256, 256, 0, stream>>>(kern, qk, wbuf);

    const size_t shmem = (size_t)LDS_FLOATS * sizeof(float);  // 301,056 B
    (void)hipFuncSetAttribute((const void*)convquad_main,
                              hipFuncAttributeMaxDynamicSharedMemorySize,
                              (int)shmem);
    const int blocks = (NTILES + WPB - 1) / WPB;  // 1954
    convquad_main<<<blocks, NTHREADS, shmem, stream>>>(x, bias, wbuf, out);
}